// Resize3DImageMask_67894843015524
// MI455X (gfx1250) — compile-verified
//
#include <hip/hip_runtime.h>
#include <stdint.h>

// input  [D0=256][D1=384][D2=384] f32
// output [1][S1=384][S0=320][S2=512] f32
#define D0 256
#define D1 384
#define D2 384
#define S0 320
#define S1 384
#define S2 512

#define TJ0      16   // output rows (dim0) per block
#define ROWS_MAX 14   // span 0.8*15=12 -> floor diff <=12, +1 nbr +1 incl = 14
#define RSTRIDE  392  // floats per staged row: col c in [-1,384] at [4+c]; 16B-aligned data, 8-bank row skew

typedef float v4f __attribute__((ext_vector_type(4)));
typedef float v2fu __attribute__((ext_vector_type(2), aligned(4)));  // unaligned-pair LDS load

__global__ __launch_bounds__(256)
void Resize3D_kernel(const float* __restrict__ in, float* __restrict__ out) {
    __shared__ float rows[ROWS_MAX * RSTRIDE];   // ~21.5 KB

    const int j0s = blockIdx.x * TJ0;    // first output dim0 index of this tile
    const int j1  = blockIdx.y;          // pass-through dim, 0..383
    const int t   = (int)threadIdx.x;    // 0..255

    // ---- staged input-row range for this tile (uniform per block) ----
    float s0f = fmaxf((j0s + 0.5f)           * 0.8f - 0.5f, 0.0f);
    float s0l = fmaxf((j0s + TJ0 - 1 + 0.5f) * 0.8f - 0.5f, 0.0f);
    const int i0_min   = (int)s0f;
    const int i0_lastb = min((int)s0l + 1, D0 - 1);
    const int nrows    = i0_lastb - i0_min + 1;          // <= ROWS_MAX

    // ---- async stage nrows x 1536B rows into LDS (CDNA5 ASYNCcnt path) ----
    // row data (cols 0..383) lands at rows[r*RSTRIDE + 4 ..], 16B-aligned chunks
    uint32_t lds_base = (uint32_t)(uintptr_t)(void*)rows;
    uint64_t gbase    = (uint64_t)(uintptr_t)(in + ((size_t)i0_min * D1 + j1) * D2);
    const int total = nrows * 96;                         // 96 x 16B chunks per row
    for (int c = t; c < total; c += 256) {
        int row   = c / 96;
        int chunk = c - row * 96;
        uint64_t gaddr = gbase + (uint32_t)row * (uint32_t)(D1 * D2 * 4) + (uint32_t)(chunk * 16);
        uint32_t laddr = lds_base + (uint32_t)(row * (RSTRIDE * 4) + 16 + chunk * 16);
        asm volatile("global_load_async_to_lds_b128 %0, %1, off"
                     :: "v"(laddr), "v"(gaddr)
                     : "memory");
    }
    // ---- edge pads (replicate col0 -> col[-1], col383 -> col[384]) via scalar global loads ----
    if (t < 2 * nrows) {
        int r = t >> 1, side = t & 1;
        float v = in[((size_t)(i0_min + r) * D1 + j1) * D2 + (side ? (D2 - 1) : 0)];
        rows[r * RSTRIDE + (side ? (4 + D2) : 3)] = v;
    }
    asm volatile("s_wait_asynccnt 0x0" ::: "memory");
    __syncthreads();

    // ---- compute: scale 384/512 = 3/4 -> constant weights per 4-output group ----
    // j2 = 4m+k reads cols 3m-1..3m+3 with weights {.125/.875, .375/.625, .625/.375, .875/.125}
    // (exact dyadic: bit-identical to the reference's fp32 coordinate math)
    const int m = t & 127;           // j2 group, 0..127
    const int p = t >> 7;            // 0 or 1: which jj parity this thread handles
    const int colbase = 4 + 3 * m;   // LDS float index of col 3m within a row

    #pragma unroll
    for (int it = 0; it < TJ0 / 2; ++it) {
        const int jj = 2 * it + p;
        const int j0 = j0s + jj;

        // dim0 (256 -> 320) coords: keep the reference's exact fp32 sequence
        float src0 = fmaxf((j0 + 0.5f) * 0.8f - 0.5f, 0.0f);
        int   i0   = (int)src0;
        float w0   = src0 - (float)i0;
        int   i0b  = min(i0 + 1, D0 - 1);
        const float* r0 = &rows[(i0  - i0_min) * RSTRIDE + colbase];
        const float* r1 = &rows[(i0b - i0_min) * RSTRIDE + colbase];

        // 5 columns, blend dim0 first; pair loads -> ds_load_2addr_b32, pair math -> v_pk ops
        v2fu A01 = *(const v2fu*)(r0 - 1);   // cols 3m-1, 3m
        v2fu A23 = *(const v2fu*)(r0 + 1);   // cols 3m+1, 3m+2
        float a4 = r0[3];                    // col 3m+3
        v2fu B01 = *(const v2fu*)(r1 - 1);
        v2fu B23 = *(const v2fu*)(r1 + 1);
        float b4 = r1[3];

        v2fu C01 = A01 + w0 * (B01 - A01);
        v2fu C23 = A23 + w0 * (B23 - A23);
        float c4 = a4 + w0 * (b4 - a4);

        // dim2 interp with exact dyadic constant weights
        v4f res;
        res.x = 0.125f * C01.x + 0.875f * C01.y;
        res.y = 0.375f * C01.y + 0.625f * C23.x;
        res.z = 0.625f * C23.x + 0.375f * C23.y;
        res.w = 0.875f * C23.y + 0.125f * c4;

        // NT store: 252MB output stream must not evict the (L2-resident, reused) input
        __builtin_nontemporal_store(res, (v4f*)(out + ((size_t)j1 * S0 + j0) * S2 + 4 * m));
    }
}

extern "C" void kernel_launch(void* const* d_in, const int* in_sizes, int n_in,
                              void* d_out, int out_size, void* d_ws, size_t ws_size,
                              hipStream_t stream) {
    (void)in_sizes; (void)n_in; (void)d_ws; (void)ws_size; (void)out_size;
    const float* in  = (const float*)d_in[0];
    float*       out = (float*)d_out;
    dim3 grid(S0 / TJ0, S1);     // 20 x 384 = 7,680 blocks, 16 output rows each
    Resize3D_kernel<<<grid, 256, 0, stream>>>(in, out);
}